// VQVAE_Tokenizer_35699768164450
// MI455X (gfx1250) — compile-verified
//
#include <hip/hip_runtime.h>
#include <hip/hip_bf16.h>
#include <math.h>

// ---------------------------------------------------------------------------
// VQ-VAE tokenizer forward for gfx1250 (MI455X).
// Flagship op: VQ nearest-codebook search as an f32 WMMA GEMM
// (v_wmma_f32_16x16x4_f32) fed from LDS, with codebook tiles DMA'd by the
// Tensor Data Mover (tensor_load_to_lds, TENSORcnt double-buffering).
// Rest of the net: small VALU conv kernels (L2-resident).
// ---------------------------------------------------------------------------

typedef __attribute__((ext_vector_type(2))) float v2f;
typedef __attribute__((ext_vector_type(8))) float v8f;
typedef unsigned int tdm_u32x4 __attribute__((ext_vector_type(4)));
typedef int          tdm_i32x8 __attribute__((ext_vector_type(8)));
typedef int          tdm_i32x4 __attribute__((ext_vector_type(4)));

#if defined(__has_builtin)
#  if __has_builtin(__builtin_amdgcn_tensor_load_to_lds)
#    define HAVE_TDM 1
#  endif
#endif
#ifndef HAVE_TDM
#  define HAVE_TDM 0
#endif

#define BB    64
#define CIN   6
#define TT    4096
#define HH    64
#define DD    128
#define KCB   1028
#define TQ    1024          // T/4
#define NTOT  (BB * TQ)     // 65536 latent vectors
#define XRN   (BB * CIN * TT)

#define TROW    130         // LDS row stride in floats (128 + 2 pad, bank-conflict free)
#define TILE_F  2112        // floats per LDS tile buffer (16*130 + slack)

// ------------------------------- init --------------------------------------
__global__ void init_accum_kernel(float* sumsq, unsigned int* hist) {
    int i = blockIdx.x * blockDim.x + threadIdx.x;
    if (i == 0) *sumsq = 0.0f;
    if (i < KCB) hist[i] = 0u;
}

// --------------------------- codebook norms --------------------------------
__global__ void codebook_norm_kernel(const float* __restrict__ cb,
                                     float* __restrict__ cnorm) {
    int k = blockIdx.x * blockDim.x + threadIdx.x;
    if (k >= KCB) return;
    const float* r = cb + (size_t)k * DD;
    float s = 0.0f;
    for (int d = 0; d < DD; ++d) s += r[d] * r[d];
    cnorm[k] = s;
}

// --------------------------- encoder convs ---------------------------------
// conv1: (B,6,4096) -> (B,32,2048), k=4 s=2 p=1, relu
__global__ __launch_bounds__(256) void enc_conv1_kernel(
    const float* __restrict__ x, const float* __restrict__ w,
    const float* __restrict__ b, float* __restrict__ out) {
    int gid = blockIdx.x * 256 + threadIdx.x;        // B*32*2048
    int t  = gid & 2047;
    int co = (gid >> 11) & 31;
    int bb = gid >> 16;
    float acc = b[co];
    int base = 2 * t - 1;
    for (int ci = 0; ci < CIN; ++ci) {
        const float* xr = x + ((size_t)bb * CIN + ci) * TT;
        const float* wr = w + (co * CIN + ci) * 4;
        #pragma unroll
        for (int j = 0; j < 4; ++j) {
            int ti = base + j;
            if (ti >= 0 && ti < TT) acc += xr[ti] * wr[j];
        }
    }
    out[gid] = fmaxf(acc, 0.0f);
}

// conv2: (B,32,2048) -> (B,64,1024), k=4 s=2 p=1, relu
__global__ __launch_bounds__(256) void enc_conv2_kernel(
    const float* __restrict__ x, const float* __restrict__ w,
    const float* __restrict__ b, float* __restrict__ out) {
    int gid = blockIdx.x * 256 + threadIdx.x;        // B*64*1024
    int t  = gid & 1023;
    int co = (gid >> 10) & 63;
    int bb = gid >> 16;
    float acc = b[co];
    int base = 2 * t - 1;
    for (int ci = 0; ci < 32; ++ci) {
        const float* xr = x + ((size_t)bb * 32 + ci) * 2048;
        const float* wr = w + (co * 32 + ci) * 4;
        #pragma unroll
        for (int j = 0; j < 4; ++j) {
            int ti = base + j;
            if (ti >= 0 && ti < 2048) acc += xr[ti] * wr[j];
        }
    }
    out[gid] = fmaxf(acc, 0.0f);
}

// conv3: (B,64,1024) -> (B,64,1024), k=3 s=1 p=1, bias, no relu
__global__ __launch_bounds__(256) void enc_conv3_kernel(
    const float* __restrict__ x, const float* __restrict__ w,
    const float* __restrict__ b, float* __restrict__ out) {
    int gid = blockIdx.x * 256 + threadIdx.x;        // B*64*1024
    int t  = gid & 1023;
    int co = (gid >> 10) & 63;
    int bb = gid >> 16;
    float acc = b[co];
    for (int ci = 0; ci < HH; ++ci) {
        const float* xr = x + ((size_t)bb * HH + ci) * TQ;
        const float* wr = w + (co * HH + ci) * 3;
        #pragma unroll
        for (int j = 0; j < 3; ++j) {
            int ti = t - 1 + j;
            if (ti >= 0 && ti < TQ) acc += xr[ti] * wr[j];
        }
    }
    out[gid] = acc;
}

// residual conv A: h = conv3x(relu(x)), 64->64, k=3 p=1, no bias
__global__ __launch_bounds__(256) void res_conv3_kernel(
    const float* __restrict__ x, const float* __restrict__ w,
    float* __restrict__ out) {
    int gid = blockIdx.x * 256 + threadIdx.x;
    int t  = gid & 1023;
    int co = (gid >> 10) & 63;
    int bb = gid >> 16;
    float acc = 0.0f;
    for (int ci = 0; ci < HH; ++ci) {
        const float* xr = x + ((size_t)bb * HH + ci) * TQ;
        const float* wr = w + (co * HH + ci) * 3;
        #pragma unroll
        for (int j = 0; j < 3; ++j) {
            int ti = t - 1 + j;
            if (ti >= 0 && ti < TQ) acc += fmaxf(xr[ti], 0.0f) * wr[j];
        }
    }
    out[gid] = acc;
}

// residual conv B fused add: x += conv1x1(relu(h)), 64->64, no bias
__global__ __launch_bounds__(256) void res_conv1_add_kernel(
    const float* __restrict__ h, const float* __restrict__ w,
    float* __restrict__ x) {
    int gid = blockIdx.x * 256 + threadIdx.x;
    int t  = gid & 1023;
    int co = (gid >> 10) & 63;
    int bb = gid >> 16;
    float acc = 0.0f;
    for (int ci = 0; ci < HH; ++ci)
        acc += fmaxf(h[((size_t)bb * HH + ci) * TQ + t], 0.0f) * w[co * HH + ci];
    x[gid] += acc;
}

// pre-VQ 1x1 conv (input relu'd here = end of res_stack), writes flat (N,128)
__global__ __launch_bounds__(256) void prevq_kernel(
    const float* __restrict__ x, const float* __restrict__ w,
    const float* __restrict__ b, float* __restrict__ flat) {
    int gid = blockIdx.x * 256 + threadIdx.x;        // N*128
    int d = gid & 127;
    int n = gid >> 7;
    int t  = n & 1023;
    int bb = n >> 10;
    float acc = b[d];
    const float* wr = w + d * HH;
    for (int ci = 0; ci < HH; ++ci)
        acc += fmaxf(x[((size_t)bb * HH + ci) * TQ + t], 0.0f) * wr[ci];
    flat[gid] = acc;
}

// ------------------------------ VQ (WMMA + TDM) ----------------------------
// One wave owns a 16-row tile of `flat` (A resident in 64 VGPRs). The block
// sweeps the codebook in 16-entry tiles, DMA'd into double-buffered LDS by
// the Tensor Data Mover (with hardware row padding -> conflict-free banks),
// and each wave runs 32x v_wmma_f32_16x16x4_f32 per tile from LDS.
__global__ __launch_bounds__(256) void vq_argmin_wmma_kernel(
    const float* __restrict__ flat, const float* __restrict__ cb,
    const float* __restrict__ cnorm, int* __restrict__ idx_out) {
    __shared__ float tiles[2][TILE_F];

    const int tid  = threadIdx.x;
    const int lane = tid & 31;
    const int wave = tid >> 5;
    const int m0   = (blockIdx.x * 8 + wave) * 16;   // 512 blocks -> 4096 tiles
    const int half = lane >> 4;                       // 0: K%4 in {0,1}, 1: {2,3}
    const int l15  = lane & 15;

#if HAVE_TDM
    const unsigned lds_base = (unsigned)(size_t)(&tiles[0][0]);
    auto issue_tile = [&](int c, int sel) {
        unsigned long long ga = (unsigned long long)(size_t)(cb + (size_t)c * 16 * DD);
        unsigned rows = (unsigned)(KCB - c * 16);
        if (rows > 16u) rows = 16u;
        // D# group 0: count=1 | lds_addr | global_addr | type=2 ("image")
        tdm_u32x4 g0;
        g0[0] = 1u;
        g0[1] = lds_base + (unsigned)sel * (unsigned)(TILE_F * sizeof(float));
        g0[2] = (unsigned)(ga & 0xFFFFFFFFull);
        g0[3] = (unsigned)((ga >> 32) & 0x01FFFFFFull) | (2u << 30);
        // D# group 1: data_size=4B, pad_enable, pad_interval=128DW, pad_amount=2DW,
        // tensor_dim0=128, tensor_dim1=rows, tile_dim0=128, tile_dim1=rows,
        // tensor_dim0_stride=128.
        tdm_i32x8 g1;
        g1[0] = (int)0x03920000u;
        g1[1] = (int)(128u << 16);
        g1[2] = (int)(rows << 16);
        g1[3] = (int)(128u << 16);
        g1[4] = (int)rows;
        g1[5] = 128;
        g1[6] = 0;
        g1[7] = 0;
        tdm_i32x4 gz = {0, 0, 0, 0};
#if defined(__clang_major__) && (__clang_major__ >= 23)
        tdm_i32x8 g5 = {0, 0, 0, 0, 0, 0, 0, 0};
        __builtin_amdgcn_tensor_load_to_lds(g0, g1, gz, gz, g5, 0);
#else
        __builtin_amdgcn_tensor_load_to_lds(g0, g1, gz, gz, 0);
#endif
    };
#else
    auto issue_tile = [&](int c, int sel) {
        // Fallback: cooperative copy, same padded layout.
        for (int e = tid; e < 16 * DD; e += 256) {
            int r = e >> 7, col = e & 127;
            int row = c * 16 + r;
            if (row >= KCB) row = KCB - 1;
            tiles[sel][r * TROW + col] = cb[(size_t)row * DD + col];
        }
    };
#endif

    // Preload A-tile: 32 k-steps of float2, fp32 16x4 A layout.
    v2f a[32];
    const float* arow = flat + (size_t)(m0 + l15) * DD + 2 * half;
    #pragma unroll
    for (int t = 0; t < 32; ++t) a[t] = *(const v2f*)(arow + 4 * t);

    float best[8];
    int   bidx[8];
    #pragma unroll
    for (int v = 0; v < 8; ++v) { best[v] = -3.4e38f; bidx[v] = 0; }

    // Prologue: DMA tile 0 into buffer 0.
#if HAVE_TDM
    if (wave == 0) issue_tile(0, 0);
    __builtin_amdgcn_s_wait_tensorcnt(0);
#else
    issue_tile(0, 0);
#endif
    __syncthreads();

    for (int c = 0; c < 65; ++c) {                    // 65*16 = 1040 >= 1028
        // Kick off the next tile into the other buffer (readers of it
        // finished before the barrier that ended the previous iteration).
#if HAVE_TDM
        if (c < 64 && wave == 0) issue_tile(c + 1, (c + 1) & 1);
#else
        if (c < 64) issue_tile(c + 1, (c + 1) & 1);
#endif
        int n  = c * 16 + l15;
        int nc = (n < KCB) ? n : (KCB - 1);
        __builtin_prefetch(cnorm + nc, 0, 0);

        const float* bp = &tiles[c & 1][l15 * TROW + 2 * half];
        v8f acc = {0.f, 0.f, 0.f, 0.f, 0.f, 0.f, 0.f, 0.f};
        #pragma unroll
        for (int t = 0; t < 32; ++t) {
            v2f bv = *(const v2f*)(bp + 4 * t);
            acc = __builtin_amdgcn_wmma_f32_16x16x4_f32(
                false, a[t], false, bv, (short)0, acc, false, false);
        }
        float cn    = cnorm[nc];
        bool  valid = (n < KCB);
        #pragma unroll
        for (int v = 0; v < 8; ++v) {
            float s = acc[v] - 0.5f * cn;
            if (valid && (s > best[v] || (s == best[v] && n < bidx[v]))) {
                best[v] = s;
                bidx[v] = n;
            }
        }
        if (c < 64) {
#if HAVE_TDM
            __builtin_amdgcn_s_wait_tensorcnt(0);   // tile c+1 landed
#endif
            __syncthreads();                        // + everyone done with buf c&1
        }
    }
    // Row M = m0 + v + 8*half is spread across the 16 lanes with equal `half`.
    #pragma unroll
    for (int off = 1; off < 16; off <<= 1) {
        #pragma unroll
        for (int v = 0; v < 8; ++v) {
            float ov = __shfl_xor(best[v], off, 32);
            int   oi = __shfl_xor(bidx[v], off, 32);
            if (ov > best[v] || (ov == best[v] && oi < bidx[v])) {
                best[v] = ov;
                bidx[v] = oi;
            }
        }
    }
    if (l15 == 0) {
        #pragma unroll
        for (int v = 0; v < 8; ++v)
            idx_out[m0 + v + 8 * half] = bidx[v];
    }
}

// --------------------- gather + commitment loss ----------------------------
__global__ __launch_bounds__(256) void gather_quant_loss_kernel(
    const int* __restrict__ idx, const float* __restrict__ cb,
    const float* __restrict__ flat, float* __restrict__ quant,
    float* __restrict__ sumsq) {
    __shared__ float red[256];
    int gid = blockIdx.x * 256 + threadIdx.x;        // N*32 threads, 4 dims each
    int n  = gid >> 5;
    int dg = (gid & 31) * 4;
    float4 q = *(const float4*)(cb + (size_t)idx[n] * DD + dg);
    float4 z = *(const float4*)(flat + (size_t)n * DD + dg);
    *(float4*)(quant + (size_t)n * DD + dg) = q;
    float d0 = q.x - z.x, d1 = q.y - z.y, d2 = q.z - z.z, d3 = q.w - z.w;
    red[threadIdx.x] = d0 * d0 + d1 * d1 + d2 * d2 + d3 * d3;
    __syncthreads();
    for (int o = 128; o > 0; o >>= 1) {
        if (threadIdx.x < o) red[threadIdx.x] += red[threadIdx.x + o];
        __syncthreads();
    }
    if (threadIdx.x == 0) atomicAdd(sumsq, red[0]);
}

__global__ void hist_kernel(const int* __restrict__ idx, unsigned int* __restrict__ hist) {
    int n = blockIdx.x * 256 + threadIdx.x;
    if (n < NTOT) atomicAdd(&hist[idx[n]], 1u);
}

// ----------------------------- decoder -------------------------------------
// dec conv1: quant flat (N,128) -> (B,64,1024), k=3 p=1, bias
__global__ __launch_bounds__(256) void dec_conv1_kernel(
    const float* __restrict__ q, const float* __restrict__ w,
    const float* __restrict__ b, float* __restrict__ out) {
    int gid = blockIdx.x * 256 + threadIdx.x;        // B*64*1024
    int t  = gid & 1023;
    int co = (gid >> 10) & 63;
    int bb = gid >> 16;
    float acc = b[co];
    #pragma unroll 1
    for (int j = 0; j < 3; ++j) {
        int ti = t - 1 + j;
        if (ti < 0 || ti >= TQ) continue;
        const float* qr = q + (size_t)(bb * TQ + ti) * DD;
        const float* wr = w + (size_t)co * DD * 3 + j;
        for (int ci = 0; ci < DD; ++ci) acc += qr[ci] * wr[ci * 3];
    }
    out[gid] = acc;
}

// transposed conv 1: (B,64,1024) -> (B,32,2048), k=4 s=2 p=1, relu(in & out)
__global__ __launch_bounds__(256) void dect1_kernel(
    const float* __restrict__ y, const float* __restrict__ wt,
    const float* __restrict__ b, float* __restrict__ out) {
    int gid = blockIdx.x * 256 + threadIdx.x;        // B*32*2048
    int t  = gid & 2047;
    int co = (gid >> 11) & 31;
    int bb = gid >> 16;
    float acc = b[co];
    for (int ci = 0; ci < HH; ++ci) {
        const float* yr = y + ((size_t)bb * HH + ci) * TQ;
        const float* wr = wt + (ci * 32 + co) * 4;
        #pragma unroll
        for (int j = 0; j < 4; ++j) {
            int num = t + 1 - j;                     // t = 2*i + j - 1
            if ((num & 1) == 0) {
                int i = num >> 1;
                if (i >= 0 && i < TQ) acc += fmaxf(yr[i], 0.0f) * wr[j];
            }
        }
    }
    out[gid] = fmaxf(acc, 0.0f);
}

// transposed conv 2: (B,32,2048) -> (B,6,4096), k=4 s=2 p=1, bias only
__global__ __launch_bounds__(256) void dect2_kernel(
    const float* __restrict__ y, const float* __restrict__ wt,
    const float* __restrict__ b, float* __restrict__ xrec) {
    int gid = blockIdx.x * 256 + threadIdx.x;        // B*6*4096
    if (gid >= XRN) return;
    int t  = gid % TT;
    int co = (gid / TT) % CIN;
    int bb = gid / (TT * CIN);
    float acc = b[co];
    for (int ci = 0; ci < 32; ++ci) {
        const float* yr = y + ((size_t)bb * 32 + ci) * 2048;
        const float* wr = wt + (ci * CIN + co) * 4;
        #pragma unroll
        for (int j = 0; j < 4; ++j) {
            int num = t + 1 - j;
            if ((num & 1) == 0) {
                int i = num >> 1;
                if (i >= 0 && i < 2048) acc += yr[i] * wr[j];
            }
        }
    }
    xrec[gid] = acc;
}

// ----------------------------- finalize ------------------------------------
__global__ __launch_bounds__(256) void finalize_kernel(
    const float* __restrict__ sumsq, const unsigned int* __restrict__ hist,
    float* __restrict__ out_loss, float* __restrict__ out_ppl) {
    __shared__ float red[256];
    float acc = 0.0f;
    for (int k = threadIdx.x; k < KCB; k += 256) {
        float p = (float)hist[k] / (float)NTOT;
        acc += p * logf(p + 1e-10f);
    }
    red[threadIdx.x] = acc;
    __syncthreads();
    for (int o = 128; o > 0; o >>= 1) {
        if (threadIdx.x < o) red[threadIdx.x] += red[threadIdx.x + o];
        __syncthreads();
    }
    if (threadIdx.x == 0) {
        *out_ppl  = expf(-red[0]);
        *out_loss = 0.25f * (*sumsq) / (float)((size_t)NTOT * DD);
    }
}

__global__ void idx_to_float_kernel(const int* __restrict__ idx, float* __restrict__ out) {
    int n = blockIdx.x * 256 + threadIdx.x;
    if (n < NTOT) out[n] = (float)idx[n];
}

// ----------------------------- launcher ------------------------------------
extern "C" void kernel_launch(void* const* d_in, const int* in_sizes, int n_in,
                              void* d_out, int out_size, void* d_ws, size_t ws_size,
                              hipStream_t stream) {
    (void)in_sizes; (void)n_in; (void)out_size; (void)ws_size;

    const float* x         = (const float*)d_in[0];
    const float* enc_w1    = (const float*)d_in[1];
    const float* enc_b1    = (const float*)d_in[2];
    const float* enc_w2    = (const float*)d_in[3];
    const float* enc_b2    = (const float*)d_in[4];
    const float* enc_w3    = (const float*)d_in[5];
    const float* enc_b3    = (const float*)d_in[6];
    const float* enc_res_w1= (const float*)d_in[7];   // (3,64,64,3)
    const float* enc_res_w2= (const float*)d_in[8];   // (3,64,64,1)
    const float* prevq_w   = (const float*)d_in[9];
    const float* prevq_b   = (const float*)d_in[10];
    const float* codebook  = (const float*)d_in[11];  // (1028,128)
    const float* dec_w1    = (const float*)d_in[12];
    const float* dec_b1    = (const float*)d_in[13];
    const float* dec_res_w1= (const float*)d_in[14];
    const float* dec_res_w2= (const float*)d_in[15];
    const float* dect1_w   = (const float*)d_in[16];
    const float* dect1_b   = (const float*)d_in[17];
    const float* dect2_w   = (const float*)d_in[18];
    const float* dect2_b   = (const float*)d_in[19];

    float* out = (float*)d_out;
    float* out_loss = out;                        // [0]
    float* out_xrec = out + 1;                    // [1 .. 1+XRN)
    float* out_ppl  = out + 1 + XRN;              // scalar
    float* out_idx  = out + 2 + XRN;              // N floats

    // Workspace carve-up (bytes)
    char* ws = (char*)d_ws;
    float*        bufA  = (float*)(ws);                              // 16 MB
    float*        bufB  = (float*)(ws + (size_t)16  * 1024 * 1024);  // 16 MB
    float*        flat  = (float*)(ws + (size_t)32  * 1024 * 1024);  // 32 MB (N*128)
    float*        quant = (float*)(ws + (size_t)64  * 1024 * 1024);  // 32 MB
    int*          idxb  = (int*)  (ws + (size_t)96  * 1024 * 1024);  // 256 KB
    float*        cnorm = (float*)(ws + (size_t)97  * 1024 * 1024);  // 4 KB+
    unsigned int* hist  = (unsigned int*)(ws + (size_t)97 * 1024 * 1024 + 8192);
    float*        sumsq = (float*)(ws + (size_t)97 * 1024 * 1024 + 16384);

    const int T256 = 256;
    const int g_act64 = (BB * HH * TQ) / T256;    // 16384 blocks (4.19M threads)

    init_accum_kernel<<<(KCB + 255) / 256, T256, 0, stream>>>(sumsq, hist);
    codebook_norm_kernel<<<(KCB + 255) / 256, T256, 0, stream>>>(codebook, cnorm);

    // ---- encoder ----
    enc_conv1_kernel<<<(BB * 32 * 2048) / T256, T256, 0, stream>>>(x, enc_w1, enc_b1, bufA);
    enc_conv2_kernel<<<g_act64, T256, 0, stream>>>(bufA, enc_w2, enc_b2, bufB);
    enc_conv3_kernel<<<g_act64, T256, 0, stream>>>(bufB, enc_w3, enc_b3, bufA);
    for (int i = 0; i < 3; ++i) {
        res_conv3_kernel<<<g_act64, T256, 0, stream>>>(bufA, enc_res_w1 + (size_t)i * HH * HH * 3, bufB);
        res_conv1_add_kernel<<<g_act64, T256, 0, stream>>>(bufB, enc_res_w2 + (size_t)i * HH * HH, bufA);
    }
    prevq_kernel<<<(NTOT * DD) / T256, T256, 0, stream>>>(bufA, prevq_w, prevq_b, flat);

    // ---- VQ: TDM-fed WMMA distance GEMM + argmin ----
    vq_argmin_wmma_kernel<<<NTOT / 16 / 8, T256, 0, stream>>>(flat, codebook, cnorm, idxb);
    gather_quant_loss_kernel<<<(NTOT * 32) / T256, T256, 0, stream>>>(idxb, codebook, flat, quant, sumsq);
    hist_kernel<<<(NTOT + 255) / 256, T256, 0, stream>>>(idxb, hist);

    // ---- decoder ----
    dec_conv1_kernel<<<g_act64, T256, 0, stream>>>(quant, dec_w1, dec_b1, bufA);
    for (int i = 0; i < 3; ++i) {
        res_conv3_kernel<<<g_act64, T256, 0, stream>>>(bufA, dec_res_w1 + (size_t)i * HH * HH * 3, bufB);
        res_conv1_add_kernel<<<g_act64, T256, 0, stream>>>(bufB, dec_res_w2 + (size_t)i * HH * HH, bufA);
    }
    dect1_kernel<<<(BB * 32 * 2048) / T256, T256, 0, stream>>>(bufA, dect1_w, dect1_b, bufB);
    dect2_kernel<<<(XRN + T256 - 1) / T256, T256, 0, stream>>>(bufB, dect2_w, dect2_b, out_xrec);

    // ---- scalars + indices ----
    finalize_kernel<<<1, T256, 0, stream>>>(sumsq, hist, out_loss, out_ppl);
    idx_to_float_kernel<<<(NTOT + 255) / 256, T256, 0, stream>>>(idxb, out_idx);
}